// MyGATConv_48352741819137
// MI455X (gfx1250) — compile-verified
//
#include <hip/hip_runtime.h>
#include <hip/hip_bf16.h>
#include <math.h>

// GATConv forward for MI455X (gfx1250, wave32).
//   0. init: out = bias broadcast, segmax = -inf, denom = 0
//   1. h = x @ W via V_WMMA_F32_16X16X4_F32; W staged in LDS via the
//      Tensor Data Mover (tensor_load_to_lds + s_wait_tensorcnt); fused
//      per-head attention logits via cross-lane reductions
//   2. edge pass 1: segment max of leaky_relu logits (float atomic-max trick)
//   3. edge pass 2: segment sum of exp(e - max)  (global_atomic_add_f32)
//   4. edge pass 3: out[dst] += alpha * h[src]   (wave per edge, 128 feats)

#define F_IN   128
#define HC     128
#define HEADS  8
#define NEG_SLOPE 0.2f

typedef __attribute__((ext_vector_type(2))) float v2f;
typedef __attribute__((ext_vector_type(8))) float v8f;
typedef __attribute__((ext_vector_type(4))) unsigned int v4u;
typedef __attribute__((ext_vector_type(8))) int v8i;
typedef __attribute__((ext_vector_type(4))) int v4i;

__device__ __forceinline__ void atomicMaxF32(float* addr, float val) {
  if (val >= 0.0f) {
    atomicMax((int*)addr, __float_as_int(val));
  } else {
    atomicMin((unsigned int*)addr, __float_as_uint(val));
  }
}

// ------------------------------------------------------------- TDM -------
// 2-D tile DMA global->LDS via the Tensor Data Mover. Descriptor per
// CDNA5 ISA: group0 = {count, lds_addr, global_addr,
// type=2}; group1 = {data_size=4B, tensor_dim0/1, tile_dim0/1, dim0_stride}.
// Rows beyond rem_rows read as zero (hardware OOB fill).
__device__ __forceinline__ void tdm_load_tile_f32(unsigned lds_addr,
                                                  const void* gptr,
                                                  unsigned dim0,      // row length (elems)
                                                  unsigned rem_rows,  // tensor rows left
                                                  unsigned tile1,     // tile rows
                                                  unsigned stride0) { // elems between rows
  unsigned long long ga = (unsigned long long)(uintptr_t)gptr;
  v4u g0;
  g0[0] = 1u;                                             // count=1 (valid)
  g0[1] = lds_addr;                                       // bits 63:32
  g0[2] = (unsigned)(ga & 0xFFFFFFFFu);                   // global_addr lo
  g0[3] = (unsigned)((ga >> 32) & 0x01FFFFFFu) | 0x80000000u;  // addr hi | type=2
  v8i g1;
  g1[0] = (int)(2u << 16);                                // data_size = 4 bytes
  g1[1] = (int)((dim0 & 0xFFFFu) << 16);                  // tensor_dim0 lo16
  g1[2] = (int)(((dim0 >> 16) & 0xFFFFu) |                // tensor_dim0 hi16
                ((rem_rows & 0xFFFFu) << 16));            // tensor_dim1 lo16
  g1[3] = (int)(((rem_rows >> 16) & 0xFFFFu) |            // tensor_dim1 hi16
                ((dim0 & 0xFFFFu) << 16));                // tile_dim0
  g1[4] = (int)(tile1 & 0xFFFFu);                         // tile_dim1 (tile_dim2=0)
  g1[5] = (int)stride0;                                   // tensor_dim0_stride lo32
  g1[6] = 0;
  g1[7] = 0;
  v4i z4 = {0, 0, 0, 0};
#if defined(__clang_major__) && (__clang_major__ >= 23)
  v8i z8 = {0, 0, 0, 0, 0, 0, 0, 0};
  __builtin_amdgcn_tensor_load_to_lds(g0, g1, z4, z4, z8, 0);
#else
  __builtin_amdgcn_tensor_load_to_lds(g0, g1, z4, z4, 0);
#endif
}

// ---------------------------------------------------------------- init ----
__global__ void gat_init(float* __restrict__ out, const float* __restrict__ bias,
                         float* __restrict__ segmax, float* __restrict__ denom,
                         int n) {
  int i = blockIdx.x * blockDim.x + threadIdx.x;
  int total = n * HC;
  if (i < total) out[i] = bias[i & (HC - 1)];
  if (i < n * HEADS) {
    segmax[i] = -__builtin_inff();
    denom[i]  = 0.0f;
  }
}

// ------------------------------------------------- GEMM + attn logits ----
// Block = 256 threads = 8 waves. Each wave computes a 16x128 stripe of
// h = x @ W with v_wmma_f32_16x16x4_f32. W staged once per block in LDS by
// the TDM (issued by wave 0 only; TENSOR ops are wave-level DMA commands).
__global__ __launch_bounds__(256)
void gat_gemm_attn(const float* __restrict__ x, const float* __restrict__ W,
                   const float* __restrict__ attn_src, const float* __restrict__ attn_dst,
                   float* __restrict__ h, float* __restrict__ a_src,
                   float* __restrict__ a_dst, int n) {
  __shared__ float Wlds[F_IN * HC];  // 64 KB

  if (threadIdx.x < 32) {  // wave 0: one DMA descriptor for the whole 64 KB
    unsigned lds_addr = (unsigned)(uintptr_t)(void*)Wlds;
    tdm_load_tile_f32(lds_addr, W, HC, F_IN, F_IN, HC);
    __builtin_amdgcn_s_wait_tensorcnt(0);  // LDS writes complete
  }
  __syncthreads();  // publish W to the other 7 waves

  const int wave = threadIdx.x >> 5;
  const int lane = threadIdx.x & 31;
  const int l16  = lane & 15;
  const int hi   = lane >> 4;  // 0: K pair {4k,4k+1}, 1: {4k+2,4k+3}
  const int rowBase = blockIdx.x * 128 + wave * 16;
  if (rowBase >= n) return;  // wave-uniform guard (EXEC stays all-ones)

  int arow = rowBase + l16;
  if (arow >= n) arow = n - 1;  // clamp for a partial last tile
  const float* __restrict__ xrow = x + (size_t)arow * F_IN;

  v8f acc[8] = {};  // 8 column tiles (== 8 heads), 16x16 f32 each

  for (int kk = 0; kk < 32; ++kk) {
    const int kb = kk * 4 + hi * 2;
    v2f a;
    a.x = xrow[kb];
    a.y = xrow[kb + 1];
#pragma unroll
    for (int ct = 0; ct < 8; ++ct) {
      v2f b;
      b.x = Wlds[kb * HC + ct * 16 + l16];
      b.y = Wlds[(kb + 1) * HC + ct * 16 + l16];
      acc[ct] = __builtin_amdgcn_wmma_f32_16x16x4_f32(
          false, a, false, b, (short)0, acc[ct], false, false);
    }
  }

  // Epilogue: lane<16: col=lane, rows r; lane>=16: col=lane-16, rows r+8.
#pragma unroll
  for (int ct = 0; ct < 8; ++ct) {
    const float as = attn_src[ct * 16 + l16];
    const float ad = attn_dst[ct * 16 + l16];
#pragma unroll
    for (int r = 0; r < 8; ++r) {
      const int m   = hi * 8 + r;
      const int row = rowBase + m;
      const bool ok = (row < n);
      if (ok) h[(size_t)row * HC + ct * 16 + l16] = acc[ct][r];
      float vs = acc[ct][r] * as;
      float vd = acc[ct][r] * ad;
#pragma unroll
      for (int mk = 8; mk >= 1; mk >>= 1) {  // stays within 16-lane halves
        vs += __shfl_xor(vs, mk, 32);
        vd += __shfl_xor(vd, mk, 32);
      }
      if (ok && l16 == 0) {
        a_src[(size_t)row * HEADS + ct] = vs;
        a_dst[(size_t)row * HEADS + ct] = vd;
      }
    }
  }
}

// ------------------------------------------------------ edge pass 1 ------
__global__ void gat_edge_max(const int* __restrict__ ei, int E, int n,
                             const float* __restrict__ a_src,
                             const float* __restrict__ a_dst,
                             float* __restrict__ segmax) {
  int t = blockIdx.x * blockDim.x + threadIdx.x;
  const int etot = E + n;
  if (t >= etot) return;
  int s, d;
  if (t < E) { s = ei[t]; d = ei[E + t]; } else { s = d = t - E; }
#pragma unroll
  for (int hh = 0; hh < HEADS; ++hh) {
    float e = a_src[(size_t)s * HEADS + hh] + a_dst[(size_t)d * HEADS + hh];
    e = (e >= 0.0f) ? e : NEG_SLOPE * e;
    atomicMaxF32(&segmax[(size_t)d * HEADS + hh], e);
  }
}

// ------------------------------------------------------ edge pass 2 ------
__global__ void gat_edge_sum(const int* __restrict__ ei, int E, int n,
                             const float* __restrict__ a_src,
                             const float* __restrict__ a_dst,
                             const float* __restrict__ segmax,
                             float* __restrict__ denom) {
  int t = blockIdx.x * blockDim.x + threadIdx.x;
  const int etot = E + n;
  if (t >= etot) return;
  int s, d;
  if (t < E) { s = ei[t]; d = ei[E + t]; } else { s = d = t - E; }
#pragma unroll
  for (int hh = 0; hh < HEADS; ++hh) {
    float e = a_src[(size_t)s * HEADS + hh] + a_dst[(size_t)d * HEADS + hh];
    e = (e >= 0.0f) ? e : NEG_SLOPE * e;
    float ex = __expf(e - segmax[(size_t)d * HEADS + hh]);
    atomicAdd(&denom[(size_t)d * HEADS + hh], ex);
  }
}

// ------------------------------------------------------ edge pass 3 ------
// One wave32 per edge: 32 lanes x float4 = 128 features; 4 lanes per head.
__global__ __launch_bounds__(256)
void gat_edge_scatter(const int* __restrict__ ei, int E, int n,
                      const float* __restrict__ a_src,
                      const float* __restrict__ a_dst,
                      const float* __restrict__ segmax,
                      const float* __restrict__ denom,
                      const float* __restrict__ h, float* __restrict__ out) {
  const int lane = threadIdx.x & 31;
  const int hh   = lane >> 2;        // head (0..7)
  const int c4   = (lane & 3) * 4;   // channel base within head
  const int totalWaves = (int)((gridDim.x * blockDim.x) >> 5);
  const int etot = E + n;
  int w = (blockIdx.x * blockDim.x + threadIdx.x) >> 5;
  for (; w < etot; w += totalWaves) {
    if (w + totalWaves < E) {  // prefetch next iteration's index stream
      __builtin_prefetch(&ei[w + totalWaves], 0, 1);
      __builtin_prefetch(&ei[E + w + totalWaves], 0, 1);
    }
    int s, d;
    if (w < E) { s = ei[w]; d = ei[E + w]; } else { s = d = w - E; }
    float e = a_src[(size_t)s * HEADS + hh] + a_dst[(size_t)d * HEADS + hh];
    e = (e >= 0.0f) ? e : NEG_SLOPE * e;
    const float alpha = __expf(e - segmax[(size_t)d * HEADS + hh]) /
                        denom[(size_t)d * HEADS + hh];
    const float4 hv = *(const float4*)(h + (size_t)s * HC + hh * 16 + c4);
    float* op = out + (size_t)d * HC + hh * 16 + c4;
    atomicAdd(op + 0, hv.x * alpha);
    atomicAdd(op + 1, hv.y * alpha);
    atomicAdd(op + 2, hv.z * alpha);
    atomicAdd(op + 3, hv.w * alpha);
  }
}

// ---------------------------------------------------------------- host ----
extern "C" void kernel_launch(void* const* d_in, const int* in_sizes, int n_in,
                              void* d_out, int out_size, void* d_ws, size_t ws_size,
                              hipStream_t stream) {
  const float* x    = (const float*)d_in[0];
  const int*   ei   = (const int*)d_in[1];
  const float* W    = (const float*)d_in[2];
  const float* as   = (const float*)d_in[3];
  const float* ad   = (const float*)d_in[4];
  const float* bias = (const float*)d_in[5];
  float* out = (float*)d_out;

  const int n = in_sizes[0] / F_IN;
  const int E = in_sizes[1] / 2;
  const int etot = E + n;

  // Workspace layout (floats): h | a_src | a_dst | segmax | denom  (~64 MB)
  float* ws     = (float*)d_ws;
  float* h      = ws;
  float* a_srcW = h      + (size_t)n * HC;
  float* a_dstW = a_srcW + (size_t)n * HEADS;
  float* segmax = a_dstW + (size_t)n * HEADS;
  float* denom  = segmax + (size_t)n * HEADS;

  {
    int blocks = (n * HC + 255) / 256;
    gat_init<<<blocks, 256, 0, stream>>>(out, bias, segmax, denom, n);
  }
  {
    int blocks = (n + 127) / 128;
    gat_gemm_attn<<<blocks, 256, 0, stream>>>(x, W, as, ad, h, a_srcW, a_dstW, n);
  }
  {
    int blocks = (etot + 255) / 256;
    gat_edge_max<<<blocks, 256, 0, stream>>>(ei, E, n, a_srcW, a_dstW, segmax);
    gat_edge_sum<<<blocks, 256, 0, stream>>>(ei, E, n, a_srcW, a_dstW, segmax, denom);
  }
  {
    int blocks = (etot + 7) / 8;  // 8 waves (edges) per 256-thread block
    gat_edge_scatter<<<blocks, 256, 0, stream>>>(ei, E, n, a_srcW, a_dstW,
                                                 segmax, denom, h, out);
  }
}